// PandaDynamics_74010876445152
// MI455X (gfx1250) — compile-verified
//
#include <hip/hip_runtime.h>
#include <hip/hip_bf16.h>
#include <math.h>

#define NJ 7
#define DT_ 0.01f
#define GRAV_ 9.81f

typedef int int4v __attribute__((ext_vector_type(4)));
typedef __attribute__((address_space(1))) int   gint_t;   // global int
typedef __attribute__((address_space(3))) int   lint_t;   // LDS int
typedef __attribute__((address_space(1))) int4v gint4_t;  // global int4
typedef __attribute__((address_space(3))) int4v lint4_t;  // LDS int4

#if __has_builtin(__builtin_amdgcn_global_load_async_to_lds_b32) && \
    __has_builtin(__builtin_amdgcn_global_load_async_to_lds_b128)
#define HAVE_ASYNC_LDS 1
#else
#define HAVE_ASYNC_LDS 0
#endif

// y = A x   (3x3 row-major)
__device__ __forceinline__ void m3v(const float* A, const float* x, float* y) {
    y[0] = fmaf(A[0], x[0], fmaf(A[1], x[1], A[2] * x[2]));
    y[1] = fmaf(A[3], x[0], fmaf(A[4], x[1], A[5] * x[2]));
    y[2] = fmaf(A[6], x[0], fmaf(A[7], x[1], A[8] * x[2]));
}
// y = A^T x
__device__ __forceinline__ void m3tv(const float* A, const float* x, float* y) {
    y[0] = fmaf(A[0], x[0], fmaf(A[3], x[1], A[6] * x[2]));
    y[1] = fmaf(A[1], x[0], fmaf(A[4], x[1], A[7] * x[2]));
    y[2] = fmaf(A[2], x[0], fmaf(A[5], x[1], A[8] * x[2]));
}
// y = a x b
__device__ __forceinline__ void cross3(const float* a, const float* b, float* y) {
    y[0] = a[1] * b[2] - a[2] * b[1];
    y[1] = a[2] * b[0] - a[0] * b[2];
    y[2] = a[0] * b[1] - a[1] * b[0];
}
// f := Xup^T f, with Xup=[[A,0],[B,A]], B = -A*S(p)  =>  B^T f2 = p x (A^T f2)
__device__ __forceinline__ void xupT_apply(const float* A, const float* p, float* f) {
    float g1[3], g2[3], pc[3];
    m3tv(A, f + 3, g2);
    m3tv(A, f, g1);
    cross3(p, g2, pc);
    f[0] = g1[0] + pc[0]; f[1] = g1[1] + pc[1]; f[2] = g1[2] + pc[2];
    f[3] = g2[0];         f[4] = g2[1];         f[5] = g2[2];
}

__global__ __launch_bounds__(256) void panda_step_kernel(
    const float* __restrict__ x, const float* __restrict__ u,
    const float* __restrict__ axes, const float* __restrict__ Rtree,
    const float* __restrict__ ptree, const float* __restrict__ Isp,
    float* __restrict__ out, int B)
{
    // ---- stage shared robot constants (357 floats) into LDS once per block ----
    __shared__ float sA[NJ * 3];
    __shared__ float sR[NJ * 9];
    __shared__ float sP[NJ * 3];
    __shared__ __align__(16) float sI[NJ * 36];
    {
        const int t = threadIdx.x;
#if HAVE_ASYNC_LDS
        // CDNA5 async global->LDS path: bypasses VGPRs, tracked by ASYNCcnt.
        if (t < 63) {
            // 63 lanes x 16B = 1008B = all of I_sp (7x36 floats), 16B aligned both sides
            __builtin_amdgcn_global_load_async_to_lds_b128(
                (gint4_t*)const_cast<float*>(Isp + 4 * t),
                (lint4_t*)&sI[4 * t], 0, 0);
            __builtin_amdgcn_global_load_async_to_lds_b32(
                (gint_t*)const_cast<float*>(Rtree + t),
                (lint_t*)&sR[t], 0, 0);
        }
        if (t < 21) {
            __builtin_amdgcn_global_load_async_to_lds_b32(
                (gint_t*)const_cast<float*>(axes + t),
                (lint_t*)&sA[t], 0, 0);
            __builtin_amdgcn_global_load_async_to_lds_b32(
                (gint_t*)const_cast<float*>(ptree + t),
                (lint_t*)&sP[t], 0, 0);
        }
#if __has_builtin(__builtin_amdgcn_s_wait_asynccnt)
        __builtin_amdgcn_s_wait_asynccnt(0);
#else
        asm volatile("s_wait_asynccnt 0x0" ::: "memory");
#endif
#else
        if (t < NJ * 3) { sA[t] = axes[t]; sP[t] = ptree[t]; }
        if (t < NJ * 9) { sR[t] = Rtree[t]; }
        for (int k = t; k < NJ * 36; k += blockDim.x) sI[k] = Isp[k];
#endif
    }
    __builtin_amdgcn_s_wait_tensorcnt(0);
    __syncthreads();

    const int b = blockIdx.x * blockDim.x + threadIdx.x;
    if (b >= B) return;

    const float QL[NJ] = {-2.9671f, -1.8326f, -2.9671f, -3.1416f, -2.9671f, -0.0873f, -2.9671f};
    const float QU[NJ] = { 2.9671f,  1.8326f,  2.9671f,  0.0f,     2.9671f,  3.8223f,  2.9671f};

    float q[NJ], qd[NJ];
#pragma unroll
    for (int i = 0; i < NJ; ++i) {
        float qi = x[(size_t)b * 14 + i];
        q[i]  = fminf(fmaxf(qi, QL[i]), QU[i]);
        qd[i] = x[(size_t)b * 14 + 7 + i];
    }

    // Xup_i = [[A,0],[B,A]] with A = R^T * R_tree[i]; B never stored (use p-cross identities)
    float XA[NJ][9];
    float fs[NJ][6];

    float vw[3] = {0.f, 0.f, 0.f}, vv[3] = {0.f, 0.f, 0.f};
    float aw[3] = {0.f, 0.f, 0.f}, av[3] = {0.f, 0.f, GRAV_};

    // ---------------- forward pass (RNEA) ----------------
#pragma unroll
    for (int i = 0; i < NJ; ++i) {
        const float a0 = sA[3 * i], a1 = sA[3 * i + 1], a2 = sA[3 * i + 2];
        const float p0 = sP[3 * i], p1 = sP[3 * i + 1], p2 = sP[3 * i + 2];
        const float avec[3] = {a0, a1, a2};
        const float pvec[3] = {p0, p1, p2};

        float s, c;
        s = sinf(q[i]); c = cosf(q[i]);
        const float omc = 1.f - c;
        float R[9];  // R = c I + s S(a) + (1-c) a a^T   (row major)
        R[0] = c + omc * a0 * a0;      R[1] = -s * a2 + omc * a0 * a1; R[2] =  s * a1 + omc * a0 * a2;
        R[3] =  s * a2 + omc * a0 * a1; R[4] = c + omc * a1 * a1;      R[5] = -s * a0 + omc * a1 * a2;
        R[6] = -s * a1 + omc * a0 * a2; R[7] =  s * a0 + omc * a1 * a2; R[8] = c + omc * a2 * a2;

        // A = R^T * E,  E = R_tree[i]
        float* A_ = XA[i];
#pragma unroll
        for (int r = 0; r < 3; ++r)
#pragma unroll
            for (int cc = 0; cc < 3; ++cc)
                A_[3 * r + cc] = fmaf(R[r],     sR[9 * i + cc],
                                 fmaf(R[3 + r], sR[9 * i + 3 + cc],
                                      R[6 + r] * sR[9 * i + 6 + cc]));

        // v = Xup v_prev + [a qd; 0];   Xup v = [A w; A (u + w x p)]
        float wxp[3], tv[3], nw[3], nv[3];
        cross3(vw, pvec, wxp);
        tv[0] = vv[0] + wxp[0]; tv[1] = vv[1] + wxp[1]; tv[2] = vv[2] + wxp[2];
        m3v(A_, vw, nw);
        nw[0] = fmaf(a0, qd[i], nw[0]); nw[1] = fmaf(a1, qd[i], nw[1]); nw[2] = fmaf(a2, qd[i], nw[2]);
        m3v(A_, tv, nv);

        // acc = Xup a_prev + crm(v) vJ
        float axp[3], ta[3], naw[3], nav[3];
        cross3(aw, pvec, axp);
        ta[0] = av[0] + axp[0]; ta[1] = av[1] + axp[1]; ta[2] = av[2] + axp[2];
        m3v(A_, aw, naw);
        m3v(A_, ta, nav);
        float wxa[3], vxa[3];
        cross3(nw, avec, wxa);
        cross3(nv, avec, vxa);
        naw[0] = fmaf(wxa[0], qd[i], naw[0]); naw[1] = fmaf(wxa[1], qd[i], naw[1]); naw[2] = fmaf(wxa[2], qd[i], naw[2]);
        nav[0] = fmaf(vxa[0], qd[i], nav[0]); nav[1] = fmaf(vxa[1], qd[i], nav[1]); nav[2] = fmaf(vxa[2], qd[i], nav[2]);

        // Iv = I_sp v ; Ia = I_sp acc  (full 6x6 matvecs, broadcast reads from LDS)
        const float v6[6] = {nw[0], nw[1], nw[2], nv[0], nv[1], nv[2]};
        const float a6[6] = {naw[0], naw[1], naw[2], nav[0], nav[1], nav[2]};
        float Iv[6], Ia[6];
#pragma unroll
        for (int r = 0; r < 6; ++r) {
            const float* Ir = &sI[36 * i + 6 * r];
            Iv[r] = fmaf(Ir[0], v6[0], fmaf(Ir[1], v6[1], fmaf(Ir[2], v6[2],
                    fmaf(Ir[3], v6[3], fmaf(Ir[4], v6[4], Ir[5] * v6[5])))));
            Ia[r] = fmaf(Ir[0], a6[0], fmaf(Ir[1], a6[1], fmaf(Ir[2], a6[2],
                    fmaf(Ir[3], a6[3], fmaf(Ir[4], a6[4], Ir[5] * a6[5])))));
        }
        // f = I acc + [w x m1 + vl x m2 ; w x m2]   (= I acc - crm(v)^T Iv)
        float c1[3], c2[3], c3[3];
        cross3(nw, &Iv[0], c1);
        cross3(nv, &Iv[3], c2);
        cross3(nw, &Iv[3], c3);
        fs[i][0] = Ia[0] + c1[0] + c2[0];
        fs[i][1] = Ia[1] + c1[1] + c2[1];
        fs[i][2] = Ia[2] + c1[2] + c2[2];
        fs[i][3] = Ia[3] + c3[0];
        fs[i][4] = Ia[4] + c3[1];
        fs[i][5] = Ia[5] + c3[2];

        vw[0] = nw[0]; vw[1] = nw[1]; vw[2] = nw[2];
        vv[0] = nv[0]; vv[1] = nv[1]; vv[2] = nv[2];
        aw[0] = naw[0]; aw[1] = naw[1]; aw[2] = naw[2];
        av[0] = nav[0]; av[1] = nav[1]; av[2] = nav[2];
    }

    // ---------------- backward pass: bias torques h ----------------
    float h[NJ];
#pragma unroll
    for (int i = NJ - 1; i >= 0; --i) {
        h[i] = fmaf(sA[3 * i], fs[i][0], fmaf(sA[3 * i + 1], fs[i][1], sA[3 * i + 2] * fs[i][2]));
        if (i > 0) {
            float g[6] = {fs[i][0], fs[i][1], fs[i][2], fs[i][3], fs[i][4], fs[i][5]};
            xupT_apply(XA[i], &sP[3 * i], g);
#pragma unroll
            for (int k = 0; k < 6; ++k) fs[i - 1][k] += g[k];
        }
    }

    // ---------------- CRBA: mass matrix ----------------
    float Mm[NJ][NJ];
    float Ic[36];
#pragma unroll
    for (int k = 0; k < 36; ++k) Ic[k] = sI[36 * (NJ - 1) + k];

#pragma unroll
    for (int i = NJ - 1; i >= 0; --i) {
        const float a0 = sA[3 * i], a1 = sA[3 * i + 1], a2 = sA[3 * i + 2];
        float F[6];
#pragma unroll
        for (int r = 0; r < 6; ++r)
            F[r] = fmaf(Ic[6 * r], a0, fmaf(Ic[6 * r + 1], a1, Ic[6 * r + 2] * a2));
        Mm[i][i] = fmaf(a0, F[0], fmaf(a1, F[1], a2 * F[2]));
#pragma unroll
        for (int j = i - 1; j >= 0; --j) {
            xupT_apply(XA[j + 1], &sP[3 * (j + 1)], F);
            float mij = fmaf(sA[3 * j], F[0], fmaf(sA[3 * j + 1], F[1], sA[3 * j + 2] * F[2]));
            Mm[i][j] = mij;
            Mm[j][i] = mij;
        }
        if (i > 0) {
            const float* A_ = XA[i];
            const float p0 = sP[3 * i], p1 = sP[3 * i + 1], p2 = sP[3 * i + 2];
            float Bm[9];  // B = -A S(p)
#pragma unroll
            for (int r = 0; r < 3; ++r) {
                Bm[3 * r + 0] = A_[3 * r + 2] * p1 - A_[3 * r + 1] * p2;
                Bm[3 * r + 1] = A_[3 * r + 0] * p2 - A_[3 * r + 2] * p0;
                Bm[3 * r + 2] = A_[3 * r + 1] * p0 - A_[3 * r + 0] * p1;
            }
            // T = Ic * Xup
            float T[36];
#pragma unroll
            for (int r = 0; r < 6; ++r) {
#pragma unroll
                for (int cc = 0; cc < 3; ++cc) {
                    T[6 * r + cc] =
                        fmaf(Ic[6 * r + 0], A_[cc],
                        fmaf(Ic[6 * r + 1], A_[3 + cc],
                        fmaf(Ic[6 * r + 2], A_[6 + cc],
                        fmaf(Ic[6 * r + 3], Bm[cc],
                        fmaf(Ic[6 * r + 4], Bm[3 + cc], Ic[6 * r + 5] * Bm[6 + cc])))));
                    T[6 * r + 3 + cc] =
                        fmaf(Ic[6 * r + 3], A_[cc],
                        fmaf(Ic[6 * r + 4], A_[3 + cc], Ic[6 * r + 5] * A_[6 + cc]));
                }
            }
            // Ic_new = I_sp[i-1] + Xup^T T
#pragma unroll
            for (int r = 0; r < 3; ++r) {
#pragma unroll
                for (int cc = 0; cc < 6; ++cc) {
                    float top =
                        fmaf(A_[r],     T[cc],
                        fmaf(A_[3 + r], T[6 + cc],
                        fmaf(A_[6 + r], T[12 + cc],
                        fmaf(Bm[r],     T[18 + cc],
                        fmaf(Bm[3 + r], T[24 + cc], Bm[6 + r] * T[30 + cc])))));
                    float bot =
                        fmaf(A_[r],     T[18 + cc],
                        fmaf(A_[3 + r], T[24 + cc], A_[6 + r] * T[30 + cc]));
                    Ic[6 * r + cc]       = sI[36 * (i - 1) + 6 * r + cc] + top;
                    Ic[6 * (r + 3) + cc] = sI[36 * (i - 1) + 6 * (r + 3) + cc] + bot;
                }
            }
        }
    }
#pragma unroll
    for (int i = 0; i < NJ; ++i) Mm[i][i] += 1e-8f;

    // ---------------- solve M qdd = u - h (Cholesky, SPD) ----------------
    float bvec[NJ];
#pragma unroll
    for (int i = 0; i < NJ; ++i) bvec[i] = u[(size_t)b * 7 + i] - h[i];

#pragma unroll
    for (int k = 0; k < NJ; ++k) {
        float d = Mm[k][k];
#pragma unroll
        for (int t = 0; t < k; ++t) d -= Mm[k][t] * Mm[k][t];
        d = sqrtf(d);
        const float inv = 1.0f / d;
        Mm[k][k] = inv;  // store reciprocal of diagonal
#pragma unroll
        for (int r = k + 1; r < NJ; ++r) {
            float e = Mm[r][k];
#pragma unroll
            for (int t = 0; t < k; ++t) e -= Mm[r][t] * Mm[k][t];
            Mm[r][k] = e * inv;
        }
    }
    float y[NJ], z[NJ];
#pragma unroll
    for (int i = 0; i < NJ; ++i) {
        float e = bvec[i];
#pragma unroll
        for (int t = 0; t < i; ++t) e -= Mm[i][t] * y[t];
        y[i] = e * Mm[i][i];
    }
#pragma unroll
    for (int i = NJ - 1; i >= 0; --i) {
        float e = y[i];
#pragma unroll
        for (int t = i + 1; t < NJ; ++t) e -= Mm[t][i] * z[t];
        z[i] = e * Mm[i][i];
    }

    // ---------------- integrate + write out ----------------
#pragma unroll
    for (int i = 0; i < NJ; ++i) {
        float qdn = fmaf(DT_, z[i], qd[i]);
        out[(size_t)b * 14 + i]     = fmaf(DT_, qdn, q[i]);
        out[(size_t)b * 14 + 7 + i] = qdn;
    }
}

extern "C" void kernel_launch(void* const* d_in, const int* in_sizes, int n_in,
                              void* d_out, int out_size, void* d_ws, size_t ws_size,
                              hipStream_t stream) {
    const float* x    = (const float*)d_in[0];
    const float* u    = (const float*)d_in[1];
    const float* axes = (const float*)d_in[2];
    const float* Rt   = (const float*)d_in[3];
    const float* pt   = (const float*)d_in[4];
    const float* Is   = (const float*)d_in[5];
    float* out = (float*)d_out;

    const int B = in_sizes[0] / 14;
    const int block = 256;
    const int grid = (B + block - 1) / block;
    hipLaunchKernelGGL(panda_step_kernel, dim3(grid), dim3(block), 0, stream,
                       x, u, axes, Rt, pt, Is, out, B);
}